// LightGCN_17119739642023
// MI455X (gfx1250) — compile-verified
//
#include <hip/hip_runtime.h>
#include <hip/hip_bf16.h>
#include <stdint.h>

#define EMB_DIM    64
#define CHUNK      256     // edges staged per block per LDS buffer
#define SPMM_BLOCK 256

// Low 32 bits of a generic pointer into LDS == LDS byte offset (shared aperture
// occupies the high 32 bits only on gfx1250).
__device__ __forceinline__ unsigned lds_off_u32(const void* p) {
    return (unsigned)(unsigned long long)p;
}

// gfx1250 async DMA: copies one dword from global (saddr + vaddr offset) into LDS at address vdst, tracked by ASYNCcnt.
__device__ __forceinline__ void async_load_b32(unsigned lds_addr, int byte_off, const void* base) {
    asm volatile("global_load_async_to_lds_b32 %0, %1, %2"
                 :: "v"(lds_addr), "v"(byte_off),
                    "s"((unsigned long long)(size_t)base)
                 : "memory");
}

__device__ __forceinline__ void wait_asynccnt0() {
    asm volatile("s_wait_asynccnt 0x0" ::: "memory");
}

// COO SpMM: y[r,:] += v * x[c,:], edges streamed through LDS with double-buffered
// async copies. 16 lanes per edge, float4 per lane (64 floats per row).
__global__ void __launch_bounds__(SPMM_BLOCK)
spmm_kernel(const float* __restrict__ x, float* __restrict__ y,
            const int* __restrict__ erow, const int* __restrict__ ecol,
            const float* __restrict__ eval, int nedges, int nchunks) {
    __shared__ int   s_row[2][CHUNK];
    __shared__ int   s_col[2][CHUNK];
    __shared__ float s_val[2][CHUNK];

    const int t = threadIdx.x;

    auto stage = [&](int buf, int cidx) {
        int e = cidx * CHUNK + t;
        if (e >= nedges) e = nedges - 1;      // clamp; duplicate source reads are harmless
        const int off = e * 4;
        async_load_b32(lds_off_u32(&s_row[buf][t]), off, erow);
        async_load_b32(lds_off_u32(&s_col[buf][t]), off, ecol);
        async_load_b32(lds_off_u32(&s_val[buf][t]), off, eval);
    };

    int c = blockIdx.x;
    if (c >= nchunks) return;

    stage(0, c);
    int buf = 0;

    const int lane16 = t & 15;    // float4 slot within the 64-float row
    const int esub   = t >> 4;    // which edge of the 16 processed per pass

    for (; c < nchunks; c += gridDim.x) {
        wait_asynccnt0();        // this wave's async stages of `buf` are done
        __syncthreads();         // every wave's stage of `buf` is done; prev buffer fully consumed

        const int cn = c + (int)gridDim.x;
        if (cn < nchunks) stage(buf ^ 1, cn);   // overlap next chunk's DMA with processing

        const int base = c * CHUNK;
        #pragma unroll 4
        for (int k = 0; k < CHUNK / 16; ++k) {
            const int ei = k * 16 + esub;
            const int e  = base + ei;
            if (e < nedges) {
                const int   r  = s_row[buf][ei];
                const int   cc = s_col[buf][ei];
                const float v  = s_val[buf][ei];
                const float4 xg = *((const float4*)(x + (size_t)cc * EMB_DIM) + lane16);
                float* dst = y + (size_t)r * EMB_DIM + (lane16 << 2);
                atomicAdd(dst + 0, v * xg.x);
                atomicAdd(dst + 1, v * xg.y);
                atomicAdd(dst + 2, v * xg.z);
                atomicAdd(dst + 3, v * xg.w);
            }
        }
        buf ^= 1;
    }
}

__global__ void axpy_kernel(float* __restrict__ acc, const float* __restrict__ add, int n4) {
    const int i = blockIdx.x * blockDim.x + threadIdx.x;
    if (i < n4) {
        float4 a = ((const float4*)acc)[i];
        const float4 b = ((const float4*)add)[i];
        a.x += b.x; a.y += b.y; a.z += b.z; a.w += b.w;
        ((float4*)acc)[i] = a;
    }
}

__global__ void axpy_scale_kernel(float* __restrict__ acc, const float* __restrict__ add,
                                  float s, int n4) {
    const int i = blockIdx.x * blockDim.x + threadIdx.x;
    if (i < n4) {
        float4 a = ((const float4*)acc)[i];
        const float4 b = ((const float4*)add)[i];
        a.x = (a.x + b.x) * s; a.y = (a.y + b.y) * s;
        a.z = (a.z + b.z) * s; a.w = (a.w + b.w) * s;
        ((float4*)acc)[i] = a;
    }
}

extern "C" void kernel_launch(void* const* d_in, const int* in_sizes, int n_in,
                              void* d_out, int out_size, void* d_ws, size_t ws_size,
                              hipStream_t stream) {
    const float* emb_user = (const float*)d_in[0];
    const float* emb_item = (const float*)d_in[1];
    const int*   erow     = (const int*)d_in[2];
    const int*   ecol     = (const int*)d_in[3];
    const float* eval     = (const float*)d_in[4];

    const int n_user_f = in_sizes[0];                 // N_USERS * 64
    const int n_item_f = in_sizes[1];                 // N_ITEMS * 64
    const int n_edges  = in_sizes[2];
    const size_t node_floats = (size_t)n_user_f + (size_t)n_item_f;
    const size_t node_bytes  = node_floats * sizeof(float);

    float* out  = (float*)d_out;
    float* bufA = (float*)d_ws;                        // cur
    float* bufB = bufA + node_floats;                  // next

    // acc = all_emb (accumulator lives in d_out); cur = all_emb
    hipMemcpyAsync(out, emb_user, (size_t)n_user_f * sizeof(float),
                   hipMemcpyDeviceToDevice, stream);
    hipMemcpyAsync(out + n_user_f, emb_item, (size_t)n_item_f * sizeof(float),
                   hipMemcpyDeviceToDevice, stream);
    hipMemcpyAsync(bufA, out, node_bytes, hipMemcpyDeviceToDevice, stream);

    const int nchunks = (n_edges + CHUNK - 1) / CHUNK;
    int spmm_grid = nchunks < 4096 ? nchunks : 4096;
    if (spmm_grid < 1) spmm_grid = 1;

    const int n4 = (int)(node_floats / 4);
    const int egrid = (n4 + 255) / 256;

    for (int layer = 0; layer < 3; ++layer) {
        hipMemsetAsync(bufB, 0, node_bytes, stream);
        spmm_kernel<<<spmm_grid, SPMM_BLOCK, 0, stream>>>(bufA, bufB, erow, ecol, eval,
                                                          n_edges, nchunks);
        if (layer < 2) {
            axpy_kernel<<<egrid, 256, 0, stream>>>(out, bufB, n4);
            float* tmp = bufA; bufA = bufB; bufB = tmp;   // cur <- next
        } else {
            axpy_scale_kernel<<<egrid, 256, 0, stream>>>(out, bufB, 0.25f, n4);
        }
    }
}